// fODFModule_32830730010972
// MI455X (gfx1250) — compile-verified
//
#include <hip/hip_runtime.h>

// ---------------------------------------------------------------------------
// Constrained spherical deconvolution on gfx1250 (MI455X), wave32.
//
// Dominant cost: per-voxel Q = P + B^T diag(mask) B  (45x45, K=362)
//   -> batched via V_WMMA_F32_16X16X4_F32, wave-per-voxel, 6 lower tiles.
// Cholesky + triangular solves: wave-local, lane-parallel, LDS column-major
//   stride 49 (conflict-free for column ops AND row ops; gcd(49,64)=1).
//
// Intra-wave LDS producer->consumer ordering is guaranteed by hardware
// (same-wave LDS ops are processed in order); we only need a *compiler*
// fence (zero instructions) so values aren't cached/reordered in registers.
// NOTE: do NOT use `volatile` here -- on generic pointers it lowers to
// system-scope flat ops with full serializing waits.
// ---------------------------------------------------------------------------

typedef float v2f __attribute__((ext_vector_type(2)));
typedef float v8f __attribute__((ext_vector_type(8)));

__device__ __forceinline__ void lds_fence() { asm volatile("" ::: "memory"); }

#define C_DIM   45
#define QS      49          // column stride for Q / P / L (floats)
#define QELEMS  (48 * QS)   // 2352 floats per 48x48 padded matrix
#define KPAD    368         // K padded to multiple of 8
#define KS      372         // BT row stride (floats), even, good banking
#define NDWI    64
#define WPW     4           // waves (voxels) per workgroup
#define TAU_F   0.1f
#define MU_F    1e-5f
#define NITERS  10

// Shared-LDS layout (floats)
#define OFF_BT   0
#define OFF_P    (48 * KS)            // 17856
#define OFF_L0   (OFF_P + QELEMS)     // 20208
#define OFF_X    (OFF_L0 + QELEMS)    // 22560  (64 x 48)
#define OFF_WAVE (OFF_X + NDWI * 48)  // 25632
// per-wave region (floats)
#define W_Q      0
#define W_MSK    QELEMS               // 2352
#define W_Z      (W_MSK + KPAD)       // 2720
#define W_Y      (W_Z + 48)           // 2768
#define W_SH     (W_Y + 48)           // 2816
#define W_DWI    (W_SH + 48)          // 2864
#define W_SIZE   (W_DWI + NDWI)       // 2928
#define LDS_FLOATS (OFF_WAVE + WPW * W_SIZE)   // 37344 floats
#define LDS_BYTES  (LDS_FLOATS * 4)            // 149376 bytes

#define WMMA_F32X4(A, B, Cacc) \
  __builtin_amdgcn_wmma_f32_16x16x4_f32(false, (A), false, (B), (short)0, (Cacc), false, false)

// ---------------------------------------------------------------------------
// Wave-local Cholesky of leading 45x45 of column-major Q (stride 49).
// ---------------------------------------------------------------------------
__device__ __forceinline__ void cholesky45(float* Q, int lane) {
#pragma unroll 1
  for (int k = 0; k < C_DIM; ++k) {
    lds_fence();                    // order vs previous k's trailing updates
    float d   = Q[k * QS + k];      // broadcast load (all lanes same addr)
    float s   = sqrtf(d);
    float inv = 1.0f / s;
    if (lane == 0) Q[k * QS + k] = s;
    int  i1 = k + 1 + lane;
    int  i2 = i1 + 32;
    bool a1 = (i1 < C_DIM), a2 = (i2 < C_DIM);
    float c1 = 0.f, c2 = 0.f;
    if (a1) { c1 = Q[k * QS + i1] * inv; Q[k * QS + i1] = c1; }
    if (a2) { c2 = Q[k * QS + i2] * inv; Q[k * QS + i2] = c2; }
    lds_fence();                    // scaled column visible before j-loop
    // Trailing update: reads column k only, writes columns k+1..44, each
    // element touched by exactly one lane -> no fences inside (pipelines).
#pragma unroll 1
    for (int j = k + 1; j < C_DIM; ++j) {
      float cj = Q[k * QS + j];     // broadcast of scaled column entry
      if (a1 && i1 >= j) Q[j * QS + i1] -= c1 * cj;
      if (a2 && i2 >= j) Q[j * QS + i2] -= c2 * cj;
    }
  }
  lds_fence();
}

// Solve L L^T x = y in place (y[0..44]); L column-major stride 49.
__device__ __forceinline__ void tri_solve45(const float* L, float* y, int lane) {
  // forward: L u = y
#pragma unroll 1
  for (int j = 0; j < C_DIM; ++j) {
    lds_fence();                    // see previous iteration's updates
    float yj = y[j] / L[j * QS + j];
    if (lane == 0) y[j] = yj;
    int i = j + 1 + lane;
    if (i < C_DIM) y[i] -= L[j * QS + i] * yj;
    i += 32;
    if (i < C_DIM) y[i] -= L[j * QS + i] * yj;
  }
  // backward: L^T x = u   (L[j][i] at [i*QS + j], stride-49 rows: no conflicts)
#pragma unroll 1
  for (int j = C_DIM - 1; j >= 0; --j) {
    lds_fence();
    float xj = y[j] / L[j * QS + j];
    if (lane == 0) y[j] = xj;
    int i = lane;
    if (i < j) y[i] -= L[i * QS + j] * xj;
    i += 32;
    if (i < j) y[i] -= L[i * QS + j] * xj;
  }
  lds_fence();
}

// mask[p] = (sum_c B[p][c]*sh[c]) < thr ? 1 : 0   (pad p>=npts -> 0)
__device__ __forceinline__ void calc_mask(const float* BT, const float* sh,
                                          float* msk, float thr,
                                          int npts, int lane) {
  lds_fence();                      // sh written cross-lane
#pragma unroll 1
  for (int p0 = 0; p0 < KPAD; p0 += 32) {
    int p = p0 + lane;
    float f = 0.f;
#pragma unroll 1
    for (int c = 0; c < C_DIM; ++c) f += BT[c * KS + p] * sh[c];
    msk[p] = (p < npts && f < thr) ? 1.0f : 0.0f;
  }
  lds_fence();
}

// ---------------------------------------------------------------------------
// Prep (1 workgroup): P = X^T X + mu*I, L0 = chol(P); both stored col-major
// padded 48x49 (zeros outside 45x45) into workspace.
// ---------------------------------------------------------------------------
__global__ void __launch_bounds__(256) csd_prep_kernel(const float* __restrict__ X,
                                                       float* __restrict__ wsP,
                                                       float* __restrict__ wsL) {
  __shared__ float Xs[NDWI * C_DIM];
  __shared__ float Pl[QELEMS];
  const int tid = threadIdx.x;
  for (int i = tid; i < NDWI * C_DIM; i += 256) Xs[i] = X[i];
  for (int i = tid; i < QELEMS; i += 256) Pl[i] = 0.f;
  __syncthreads();
  for (int idx = tid; idx < C_DIM * C_DIM; idx += 256) {
    int i = idx % C_DIM, j = idx / C_DIM;
    float s = 0.f;
    for (int n = 0; n < NDWI; ++n) s += Xs[n * C_DIM + i] * Xs[n * C_DIM + j];
    if (i == j) s += MU_F;
    Pl[j * QS + i] = s;
  }
  __syncthreads();
  for (int i = tid; i < QELEMS; i += 256) wsP[i] = Pl[i];
  // block-parallel Cholesky of Pl
  for (int k = 0; k < C_DIM; ++k) {
    float d = Pl[k * QS + k];
    __syncthreads();                 // everyone read old diag
    float s = sqrtf(d), inv = 1.0f / s;
    if (tid == 0) Pl[k * QS + k] = s;
    for (int i = k + 1 + tid; i < C_DIM; i += 256) Pl[k * QS + i] *= inv;
    __syncthreads();                 // column scaled
    int tx = tid & 31, g = tid >> 5;  // 8 groups of 32
    for (int j = k + 1 + g; j < C_DIM; j += 8) {
      float cj = Pl[k * QS + j];
      for (int i = j + tx; i < C_DIM; i += 32)
        Pl[j * QS + i] -= Pl[k * QS + i] * cj;
    }
    __syncthreads();                 // trailing done before next diag read
  }
  for (int idx = tid; idx < QELEMS; idx += 256) {
    int j = idx / QS, i = idx % QS;
    wsL[idx] = (i < C_DIM && j < C_DIM && i >= j) ? Pl[idx] : 0.f;
  }
}

// ---------------------------------------------------------------------------
// Main kernel: wave-per-voxel, WPW voxels per workgroup.
// ---------------------------------------------------------------------------
__global__ void __launch_bounds__(WPW * 32) csd_solve_kernel(
    const float* __restrict__ dwi, const float* __restrict__ X,
    const float* __restrict__ Breg, const float* __restrict__ wsP,
    const float* __restrict__ wsL, float* __restrict__ out,
    int V, int NPTS) {
  extern __shared__ float smem[];
  float* BT = smem + OFF_BT;
  float* Ps = smem + OFF_P;
  float* Ls = smem + OFF_L0;
  float* Xs = smem + OFF_X;

  const int tid  = threadIdx.x;
  const int lane = tid & 31;
  const int w    = tid >> 5;

  // --- cooperative shared fill -------------------------------------------
  for (int idx = tid; idx < 48 * KS; idx += WPW * 32) {
    int c = idx / KS, k = idx - c * KS;
    float v = 0.f;
    if (c < C_DIM && k < NPTS) v = Breg[k * C_DIM + c];
    BT[idx] = v;
  }
  for (int idx = tid; idx < QELEMS; idx += WPW * 32) { Ps[idx] = wsP[idx]; Ls[idx] = wsL[idx]; }
  for (int idx = tid; idx < NDWI * 48; idx += WPW * 32) {
    int n = idx / 48, c = idx - n * 48;
    Xs[idx] = (c < C_DIM) ? X[n * C_DIM + c] : 0.f;
  }
  __syncthreads();

  float* Wv  = smem + OFF_WAVE + w * W_SIZE;
  float* Qc  = Wv + W_Q;
  float* msk = Wv + W_MSK;
  float* zs  = Wv + W_Z;
  float* ys  = Wv + W_Y;
  float* shs = Wv + W_SH;
  float* dsb = Wv + W_DWI;

  const int v = blockIdx.x * WPW + w;
  if (v < V) {
    // --- stage dwi row, z = X^T d ----------------------------------------
    dsb[lane]      = dwi[(size_t)v * NDWI + lane];
    dsb[lane + 32] = dwi[(size_t)v * NDWI + lane + 32];
    lds_fence();
    for (int cc = 0; cc < 2; ++cc) {
      int c = lane + 32 * cc;
      if (c < C_DIM) {
        float a = 0.f;
#pragma unroll 1
        for (int n = 0; n < NDWI; ++n) a += Xs[n * 48 + c] * dsb[n];
        zs[c] = a;
      } else if (c < 48) {
        zs[c] = 0.f;
      }
    }
    lds_fence();
    // --- initial solve with shared L0, threshold, initial mask -----------
    for (int i = lane; i < 48; i += 32) ys[i] = zs[i];
    tri_solve45(Ls, ys, lane);
    for (int i = lane; i < 48; i += 32) shs[i] = (i < C_DIM) ? ys[i] : 0.f;
    lds_fence();
    const float thr = Breg[0] * shs[0] * TAU_F;
    calc_mask(BT, shs, msk, thr, NPTS, lane);

    const int   l15   = lane & 15;
    const int   half8 = (lane & 16) ? 8 : 0;
    const int   koff  = (lane & 16) ? 2 : 0;
    const float* bt0 = &BT[(0  + l15) * KS];
    const float* bt1 = &BT[(16 + l15) * KS];
    const float* bt2 = &BT[(32 + l15) * KS];

    // --- main iterations --------------------------------------------------
#pragma unroll 1
    for (int it = 0; it < NITERS; ++it) {
      // Q = P + (mask .* B)^T B : 6 lower-triangle 16x16 tiles via WMMA f32
      v8f acc0, acc1, acc2, acc3, acc4, acc5;
#pragma unroll
      for (int r = 0; r < 8; ++r) {
        int row = r + half8;
        acc0[r] = Ps[(l15) * QS + row];
        acc1[r] = Ps[(l15) * QS + 16 + row];
        acc2[r] = Ps[(16 + l15) * QS + 16 + row];
        acc3[r] = Ps[(l15) * QS + 32 + row];
        acc4[r] = Ps[(16 + l15) * QS + 32 + row];
        acc5[r] = Ps[(32 + l15) * QS + 32 + row];
      }
      lds_fence();                  // mask written cross-lane by calc_mask
#pragma unroll 1
      for (int kk = 0; kk < KPAD; kk += 4) {
        int kb = kk + koff;
        v2f m2 = *(const v2f*)(msk + kb);   // ds_load_b64 (broadcast per half)
        v2f f0 = *(const v2f*)(bt0 + kb);   // shared fragment: B operand
        v2f f1 = *(const v2f*)(bt1 + kb);
        v2f f2 = *(const v2f*)(bt2 + kb);
        v2f a0 = { f0.x * m2.x, f0.y * m2.y };  // masked fragment: A operand
        v2f a1 = { f1.x * m2.x, f1.y * m2.y };
        v2f a2 = { f2.x * m2.x, f2.y * m2.y };
        acc0 = WMMA_F32X4(a0, f0, acc0);    // tile (0,0)
        acc1 = WMMA_F32X4(a1, f0, acc1);    // tile (1,0)
        acc2 = WMMA_F32X4(a1, f1, acc2);    // tile (1,1)
        acc3 = WMMA_F32X4(a2, f0, acc3);    // tile (2,0)
        acc4 = WMMA_F32X4(a2, f1, acc4);    // tile (2,1)
        acc5 = WMMA_F32X4(a2, f2, acc5);    // tile (2,2)
      }
#pragma unroll
      for (int r = 0; r < 8; ++r) {
        int row = r + half8;
        Qc[(l15) * QS + row]           = acc0[r];
        Qc[(l15) * QS + 16 + row]      = acc1[r];
        Qc[(16 + l15) * QS + 16 + row] = acc2[r];
        Qc[(l15) * QS + 32 + row]      = acc3[r];
        Qc[(16 + l15) * QS + 32 + row] = acc4[r];
        Qc[(32 + l15) * QS + 32 + row] = acc5[r];
      }
      // factor + solve (wave-local, no barriers)
      cholesky45(Qc, lane);
      for (int i = lane; i < 48; i += 32) ys[i] = zs[i];
      tri_solve45(Qc, ys, lane);
      for (int i = lane; i < 48; i += 32) shs[i] = (i < C_DIM) ? ys[i] : 0.f;
      if (it != NITERS - 1) calc_mask(BT, shs, msk, thr, NPTS, lane);
    }
    lds_fence();
    // --- output -----------------------------------------------------------
    for (int c = lane; c < C_DIM; c += 32) out[(size_t)v * C_DIM + c] = shs[c];
  }
}

// ---------------------------------------------------------------------------
extern "C" void kernel_launch(void* const* d_in, const int* in_sizes, int n_in,
                              void* d_out, int out_size, void* d_ws, size_t ws_size,
                              hipStream_t stream) {
  (void)n_in; (void)out_size; (void)ws_size;
  const float* dwi  = (const float*)d_in[0];
  const float* X    = (const float*)d_in[1];
  const float* Breg = (const float*)d_in[2];
  const int V    = in_sizes[0] / NDWI;   // 16384
  const int NPTS = in_sizes[2] / C_DIM;  // 362

  float* wsP = (float*)d_ws;
  float* wsL = wsP + QELEMS;

  csd_prep_kernel<<<1, 256, 0, stream>>>(X, wsP, wsL);
  csd_solve_kernel<<<(V + WPW - 1) / WPW, WPW * 32, LDS_BYTES, stream>>>(
      dwi, X, Breg, wsP, wsL, (float*)d_out, V, NPTS);
}